// GeometryLayer_17214228922754
// MI455X (gfx1250) — compile-verified
//
#include <hip/hip_runtime.h>
#include <math.h>

// Problem constants (from reference)
#define BS   4
#define LL   4800      // H0*W0
#define SS   4800      // H1*W1
#define AA   32
#define MM   512
#define CC   256
#define GW   80        // W0 == W1
#define KTOT 320       // C + 2A
#define KGEO 64        // 2A
#define RS   20        // row splits for column-stats pass
#define RPS  240       // LL / RS
#define WPAD 322       // LDS stride for W tile (320 % 64 == 0 would be a full bank conflict)

typedef __attribute__((ext_vector_type(2))) float v2f;
typedef __attribute__((ext_vector_type(4))) float v4f;
typedef __attribute__((ext_vector_type(8))) float v8f;

// ---------------- tiny weight MLP (3->3 leaky ->3, residual, ->1, tanh) ----------------
__device__ __forceinline__ float weight_mlp(float mx, float sd, float en,
                                            const float* __restrict__ W1, const float* __restrict__ b1,
                                            const float* __restrict__ W2, const float* __restrict__ b2,
                                            const float* __restrict__ hW, const float* __restrict__ hb) {
    float f0 = mx, f1 = sd, f2 = en;
    float h[3], r[3];
    #pragma unroll
    for (int j = 0; j < 3; ++j) {
        float v = b1[j] + f0 * W1[0 * 3 + j] + f1 * W1[1 * 3 + j] + f2 * W1[2 * 3 + j];
        h[j] = (v >= 0.0f) ? v : 0.01f * v;           // LeakyReLU(0.01)
    }
    #pragma unroll
    for (int j = 0; j < 3; ++j)
        r[j] = b2[j] + h[0] * W2[0 * 3 + j] + h[1] * W2[1 * 3 + j] + h[2] * W2[2 * 3 + j];
    float y = hb[0] + (f0 + r[0]) * hW[0] + (f1 + r[1]) * hW[1] + (f2 + r[2]) * hW[2];
    return tanhf(y);
}

// ---------------- top-A selection + anchor coords ----------------
__global__ void topk_anchors_kernel(const float* __restrict__ mconf,
                                    const int* __restrict__ i_ids,
                                    const int* __restrict__ j_ids,
                                    float* __restrict__ anch /* [2][BS][AA][2] */) {
    __shared__ float sval[MM];
    __shared__ float rv[256];
    __shared__ int   ri[256];
    const int b = blockIdx.x, t = threadIdx.x;
    sval[t]       = mconf[b * MM + t];
    sval[t + 256] = mconf[b * MM + t + 256];
    __syncthreads();
    for (int a = 0; a < AA; ++a) {
        float v1 = sval[t], v2 = sval[t + 256];
        if (v2 > v1) { rv[t] = v2; ri[t] = t + 256; } else { rv[t] = v1; ri[t] = t; }
        __syncthreads();
        for (int s = 128; s > 0; s >>= 1) {
            if (t < s) {
                if (rv[t + s] > rv[t] || (rv[t + s] == rv[t] && ri[t + s] < ri[t])) {
                    rv[t] = rv[t + s]; ri[t] = ri[t + s];
                }
            }
            __syncthreads();
        }
        if (t == 0) {
            int best = ri[0];
            int ii = i_ids[b * MM + best];
            int jj = j_ids[b * MM + best];
            float* a0 = anch + ((0 * BS + b) * AA + a) * 2;
            float* a1 = anch + ((1 * BS + b) * AA + a) * 2;
            a0[0] = (float)(ii / GW); a0[1] = (float)(ii % GW);
            a1[0] = (float)(jj / GW); a1[1] = (float)(jj % GW);
            sval[best] = -1.0f;   // exclude (mconf in [0,1))
        }
        __syncthreads();
    }
}

// ---------------- axis-2 (row) stats + fused MLP -> w0 ----------------
// grid: BS*LL blocks of 256; one block per conf row; b128 strided reads.
__global__ void row_stats_kernel(const float* __restrict__ conf,
                                 const float* __restrict__ W1, const float* __restrict__ b1,
                                 const float* __restrict__ W2, const float* __restrict__ b2,
                                 const float* __restrict__ hW, const float* __restrict__ hb,
                                 float* __restrict__ w0) {
    __shared__ float smx[256], ssm[256], ss2[256], sen[256];
    const int b = blockIdx.x / LL, i = blockIdx.x % LL, t = threadIdx.x;
    const float* row = conf + ((size_t)b * LL + i) * SS;
    float mx = -3.4e38f, sm = 0.f, s2 = 0.f, en = 0.f;
    for (int j0 = t * 4; j0 < SS; j0 += 1024) {          // SS % 4 == 0, rows 16B aligned
        v4f c4 = *(const v4f*)(row + j0);
        #pragma unroll
        for (int c = 0; c < 4; ++c) {
            float v = c4[c];
            mx = fmaxf(mx, v);
            sm += v; s2 += v * v;
            float cc = fmaxf(v, 1e-5f);
            en += -cc * __logf(cc);
        }
    }
    smx[t] = mx; ssm[t] = sm; ss2[t] = s2; sen[t] = en;
    __syncthreads();
    for (int s = 128; s > 0; s >>= 1) {
        if (t < s) {
            smx[t] = fmaxf(smx[t], smx[t + s]);
            ssm[t] += ssm[t + s]; ss2[t] += ss2[t + s]; sen[t] += sen[t + s];
        }
        __syncthreads();
    }
    if (t == 0) {
        float s = ssm[0], q = ss2[0];
        float var = fmaxf(q - s * s / (float)SS, 0.0f) / (float)(SS - 1);
        w0[b * LL + i] = weight_mlp(smx[0], sqrtf(var), sen[0] / (float)SS,
                                    W1, b1, W2, b2, hW, hb);
    }
}

// ---------------- axis-1 (column) stats: partial pass ----------------
// grid: (5, RS, BS); thread owns 4 columns (one b128 per row), walks RPS rows.
__global__ void col_part_kernel(const float* __restrict__ conf,
                                float* __restrict__ part /* [BS][RS][SS][4] */) {
    const int jt = blockIdx.x * 256 + threadIdx.x;        // float4 column index
    if (jt >= SS / 4) return;
    const int j = jt * 4;
    const int y = blockIdx.y, b = blockIdx.z;
    const float* base = conf + (size_t)b * LL * SS + (size_t)y * RPS * SS + j;
    float mx[4], sm[4], s2[4], en[4];
    #pragma unroll
    for (int c = 0; c < 4; ++c) { mx[c] = -3.4e38f; sm[c] = 0.f; s2[c] = 0.f; en[c] = 0.f; }
    for (int i = 0; i < RPS; ++i) {
        const float* pp = base + (size_t)i * SS;
        if ((i & 15) == 0)                                 // hide stride-19.2KB DRAM latency
            __builtin_prefetch(pp + (size_t)16 * SS, 0, 0);
        v4f c4 = *(const v4f*)pp;
        #pragma unroll
        for (int c = 0; c < 4; ++c) {
            float v = c4[c];
            mx[c] = fmaxf(mx[c], v);
            sm[c] += v; s2[c] += v * v;
            float cc = fmaxf(v, 1e-5f);
            en[c] += -cc * __logf(cc);
        }
    }
    #pragma unroll
    for (int c = 0; c < 4; ++c) {
        float* p = part + (((size_t)b * RS + y) * SS + j + c) * 4;
        p[0] = mx[c]; p[1] = sm[c]; p[2] = s2[c]; p[3] = en[c];
    }
}

// ---------------- column stats finalize + fused MLP -> w1 ----------------
__global__ void col_fin_kernel(const float* __restrict__ part,
                               const float* __restrict__ W1, const float* __restrict__ b1,
                               const float* __restrict__ W2, const float* __restrict__ b2,
                               const float* __restrict__ hW, const float* __restrict__ hb,
                               float* __restrict__ w1) {
    const int idx = blockIdx.x * 256 + threadIdx.x;       // [0, BS*SS)
    const int b = idx / SS, j = idx % SS;
    float mx = -3.4e38f, sm = 0.f, s2 = 0.f, en = 0.f;
    for (int y = 0; y < RS; ++y) {
        v4f p = *(const v4f*)(part + (((size_t)b * RS + y) * SS + j) * 4);
        mx = fmaxf(mx, p[0]); sm += p[1]; s2 += p[2]; en += p[3];
    }
    float var = fmaxf(s2 - sm * sm / (float)LL, 0.0f) / (float)(LL - 1);
    w1[idx] = weight_mlp(mx, sqrtf(var), en / (float)LL, W1, b1, W2, b2, hW, hb);
}

// ---------------- geo features: one wave32 per grid point (A==32 lanes) ----------------
__global__ void geo_kernel(const float* __restrict__ anch /* [2][BS][AA][2] */,
                           const float* __restrict__ warr /* w0 | w1 */,
                           const float* __restrict__ geo_W /* [96][64] */,
                           const float* __restrict__ geo_b /* [64] */,
                           float* __restrict__ geoB) {
    __shared__ float g[8][3 * AA];
    const int side = blockIdx.y;
    const int wave = threadIdx.x >> 5, lane = threadIdx.x & 31;
    const int p = blockIdx.x * 8 + wave;                  // [0, BS*LL)
    const int b = p / LL, n = p % LL;

    const float* ap = anch + ((size_t)side * BS * AA + (size_t)b * AA + lane) * 2;
    float px = (float)(n / GW), py = (float)(n % GW);
    float dx = px - ap[0], dy = py - ap[1];
    float dd = sqrtf(dx * dx + dy * dy);
    g[wave][3 * lane + 0] = fminf(fmaxf(dx, -100.f), 100.f) * 0.01f;
    g[wave][3 * lane + 1] = fminf(fmaxf(dy, -100.f), 100.f) * 0.01f;
    g[wave][3 * lane + 2] = fminf(dd, 100.f) * 0.01f;
    __syncthreads();

    float wv = warr[side * BS * LL + p];
    float a0 = geo_b[lane], a1 = geo_b[lane + 32];
    #pragma unroll 8
    for (int k = 0; k < 3 * AA; ++k) {
        float gv = g[wave][k];
        a0 += gv * geo_W[k * KGEO + lane];
        a1 += gv * geo_W[k * KGEO + lane + 32];
    }
    float* orow = geoB + ((size_t)side * BS * LL + p) * KGEO;
    orow[lane]      = a0 * wv;
    orow[lane + 32] = a1 * wv;
}

// ---------------- merge GEMM via V_WMMA_F32_16X16X4_F32 ----------------
// rows = 2*BS*LL = 38400 (out0|out1), cols = 256, K = 320 (virtual concat feat|geo).
// Block: 256 thr = 8 waves, one 16-col tile; W tile (320x16) staged in LDS, transposed &
// padded so a B fragment is one ds_load_b64. Each wave computes a 32x16 output (two
// accumulators) reusing the B fragment for two WMMAs per K-step.
__global__ void merge_gemm_kernel(const float* __restrict__ feat0,
                                  const float* __restrict__ feat1,
                                  const float* __restrict__ geoB,
                                  const float* __restrict__ mW /* [320][256] */,
                                  const float* __restrict__ mb /* [256] */,
                                  float* __restrict__ out) {
    __shared__ float Wt[16 * WPAD];                       // Wt[n][k], k contiguous
    const int col0 = blockIdx.y * 16;

    // cooperative W-tile load: 5120 elements, 20 iters of 256 threads
    #pragma unroll
    for (int it = 0; it < 20; ++it) {
        int idx = threadIdx.x + it * 256;                 // idx = k*16 + n
        int k = idx >> 4, n = idx & 15;
        Wt[n * WPAD + k] = mW[(size_t)k * CC + col0 + n];
    }
    __syncthreads();

    const int wave = threadIdx.x >> 5, lane = threadIdx.x & 31;
    const int rowGrp = blockIdx.x * 8 + wave;             // 32-row group
    const int row0 = rowGrp * 32;
    const int m  = lane & 15;                             // A row within tile / B & D col
    const int kb = (lane >> 4) << 1;                      // K sub-offset: 0 or 2

    const int rA0  = row0 + m;                            // concat rows (side-uniform: 19200%32==0)
    const int side = rA0 / (BS * LL);
    const int rloc = rA0 - side * (BS * LL);
    const float* fbase = side ? feat1 : feat0;
    const float* featRow0 = fbase + (size_t)rloc * CC;
    const float* featRow1 = featRow0 + (size_t)16 * CC;
    const float* geoRow0  = geoB + (size_t)rA0 * KGEO;
    const float* geoRow1  = geoRow0 + (size_t)16 * KGEO;
    const float* wrow = Wt + m * WPAD;

    v8f acc0 = {}, acc1 = {};
    for (int k0 = 0; k0 < KTOT; k0 += 4) {
        const int k = k0 + kb;
        v2f bfr = *(const v2f*)(wrow + k);                // ds_load_b64 (shared per block)
        v2f a0  = (k < CC) ? *(const v2f*)(featRow0 + k)
                           : *(const v2f*)(geoRow0 + (k - CC));
        v2f a1  = (k < CC) ? *(const v2f*)(featRow1 + k)
                           : *(const v2f*)(geoRow1 + (k - CC));
        acc0 = __builtin_amdgcn_wmma_f32_16x16x4_f32(false, a0, false, bfr,
                                                     (short)0, acc0, false, false);
        acc1 = __builtin_amdgcn_wmma_f32_16x16x4_f32(false, a1, false, bfr,
                                                     (short)0, acc1, false, false);
    }

    const int hi = lane >> 4;
    const float bias = mb[col0 + m];
    #pragma unroll
    for (int r = 0; r < 8; ++r) {                         // D layout: M = r + 8*(lane>=16)
        int gr0 = row0 + r + 8 * hi;
        out[(size_t)gr0 * CC + col0 + m]        = acc0[r] + bias;
        out[(size_t)(gr0 + 16) * CC + col0 + m] = acc1[r] + bias;
    }
}

// ---------------- launch ----------------
extern "C" void kernel_launch(void* const* d_in, const int* in_sizes, int n_in,
                              void* d_out, int out_size, void* d_ws, size_t ws_size,
                              hipStream_t stream) {
    const float* feat0 = (const float*)d_in[0];
    const float* feat1 = (const float*)d_in[1];
    const float* conf  = (const float*)d_in[2];
    const float* mconf = (const float*)d_in[3];
    const int*   i_ids = (const int*)d_in[4];
    const int*   j_ids = (const int*)d_in[5];
    const float* geo_W = (const float*)d_in[6];
    const float* geo_b = (const float*)d_in[7];
    const float* rW1   = (const float*)d_in[8];
    const float* rb1   = (const float*)d_in[9];
    const float* rW2   = (const float*)d_in[10];
    const float* rb2   = (const float*)d_in[11];
    const float* hW    = (const float*)d_in[12];
    const float* hb    = (const float*)d_in[13];
    const float* mW    = (const float*)d_in[14];
    const float* mb    = (const float*)d_in[15];
    float* out = (float*)d_out;

    // workspace carve-up (all written before read)
    float* anch = (float*)d_ws;                              // 2*BS*AA*2        = 512
    float* warr = anch + 512;                                // 2*BS*LL          = 38400
    float* colp = warr + 2 * BS * LL;                        // BS*RS*SS*4       = 1,536,000
    float* geoB = colp + (size_t)BS * RS * SS * 4;           // 2*BS*LL*KGEO     = 2,457,600

    topk_anchors_kernel<<<BS, 256, 0, stream>>>(mconf, i_ids, j_ids, anch);
    row_stats_kernel<<<BS * LL, 256, 0, stream>>>(conf, rW1, rb1, rW2, rb2, hW, hb, warr);
    col_part_kernel<<<dim3((SS / 4 + 255) / 256, RS, BS), 256, 0, stream>>>(conf, colp);
    col_fin_kernel<<<(BS * SS) / 256, 256, 0, stream>>>(colp, rW1, rb1, rW2, rb2, hW, hb,
                                                        warr + BS * LL);
    geo_kernel<<<dim3(BS * LL / 8, 2), 256, 0, stream>>>(anch, warr, geo_W, geo_b, geoB);
    merge_gemm_kernel<<<dim3((2 * BS * LL) / 32 / 8, CC / 16), 256, 0, stream>>>(
        feat0, feat1, geoB, mW, mb, out);
}